// ConvLocal2d_45586782880102
// MI455X (gfx1250) — compile-verified
//
#include <hip/hip_runtime.h>
#include <hip/hip_bf16.h>

typedef __attribute__((ext_vector_type(2))) float v2f;
typedef __attribute__((ext_vector_type(8))) float v8f;

// Problem constants (from reference): B=8, C=32, H=W=64, O=32, KH=KW=3, PAD=1
// K = C*KH*KW = 288, locations = 64*64 = 4096, per-location weight row = 288 f32.

#define WAVES_PER_BLOCK 4

__global__ __launch_bounds__(128) void ConvLocal2d_wmma_f32_kernel(
    const float* __restrict__ x,      // (8,32,64,64)
    const float* __restrict__ w,      // (64,64,32,32,3,3) -> per-loc 32x288, row-major, col = c*9+kw*3+kh
    const float* __restrict__ bias,   // (32,)
    float* __restrict__ out)          // (8,32,64,64)
{
    __shared__ float patch[WAVES_PER_BLOCK][8][288];

    const int lane = threadIdx.x & 31;
    const int wave = threadIdx.x >> 5;
    const int loc  = blockIdx.x * WAVES_PER_BLOCK + wave;   // 0..4095
    const int i = loc >> 6;
    const int j = loc & 63;

    // ---------------------------------------------------------------
    // Build the 8x288 patch for this location in LDS, stored directly in
    // weight order k' = c*9 + kw*3 + kh, holding x[b, c, i+kh-1, j+kw-1]
    // (this absorbs the einsum's (kh,kw) <-> (kw,kh) transpose for free).
    // ---------------------------------------------------------------
    float (*p)[288] = patch[wave];
    for (int idx = lane; idx < 256; idx += 32) {
        const int b = idx >> 5;
        const int c = idx & 31;
        const float* xb = x + (size_t)(b * 32 + c) * 4096;   // (64,64) plane
        float* dst = &p[b][c * 9];
        #pragma unroll
        for (int kw = 0; kw < 3; ++kw) {
            const int col = j + kw - 1;
            const bool cok = (unsigned)col < 64u;
            #pragma unroll
            for (int kh = 0; kh < 3; ++kh) {
                const int row = i + kh - 1;
                float v = (cok && (unsigned)row < 64u) ? xb[row * 64 + col] : 0.0f;
                dst[kw * 3 + kh] = v;
            }
        }
    }
    __syncthreads();

    // ---------------------------------------------------------------
    // Per-wave GEMM: out[o, b] = W_loc[o, :] . patch[b, :]
    //   A (16x4 f32): lane<16 holds row M=lane, K = k..k+1; lane>=16 -> K = k+2..k+3
    //   B (4x16 f32): lane<16 holds row K=k   (N=lane);     lane>=16 -> rows k+2/k+3
    // A streams straight from HBM (each lane walks one contiguous weight row).
    // ---------------------------------------------------------------
    const int mrow   = lane & 15;
    const int colOff = (lane >> 4) * 2;
    const float* wloc = w + (size_t)loc * (32 * 288);
    const float* a0p  = wloc + mrow * 288 + colOff;          // o-tile 0 (o = mrow)
    const float* a1p  = a0p + 16 * 288;                      // o-tile 1 (o = 16 + mrow)
    const float* bp   = &p[lane & 7][colOff];                // batch row (lanes 8..15 dup, cols discarded)

    __builtin_prefetch(a0p, 0, 0);   // global_prefetch_b8
    __builtin_prefetch(a1p, 0, 0);

    // Accumulators pre-loaded with bias: D VGPR r, lane -> (o = r + 8*(lane>=16), b = lane&15)
    const int hi8 = (lane >> 4) * 8;
    v8f c0, c1;
    #pragma unroll
    for (int r = 0; r < 8; ++r) {
        c0[r] = bias[r + hi8];
        c1[r] = bias[16 + r + hi8];
    }

    #pragma unroll 4
    for (int k = 0; k < 288; k += 4) {
        v2f a0 = *(const v2f*)(a0p + k);
        v2f a1 = *(const v2f*)(a1p + k);
        v2f bb = *(const v2f*)(bp + k);
        // D = A*B + C, exact fp32 path (matches fp32 reference semantics)
        c0 = __builtin_amdgcn_wmma_f32_16x16x4_f32(false, a0, false, bb, (short)0, c0, false, false);
        c1 = __builtin_amdgcn_wmma_f32_16x16x4_f32(false, a1, false, bb, (short)0, c1, false, false);
    }

    // ---------------------------------------------------------------
    // Store: valid batch columns are N = 0..7 (lanes 0-7 and 16-23).
    // out[b, o, i, j] = (b*32 + o)*4096 + loc
    // ---------------------------------------------------------------
    const int b = lane & 15;
    if (b < 8) {
        float* op = out + (size_t)b * (32 * 4096) + loc;
        #pragma unroll
        for (int r = 0; r < 8; ++r) {
            const int o0 = r + hi8;
            op[(size_t)o0 * 4096]        = c0[r];
            op[(size_t)(o0 + 16) * 4096] = c1[r];
        }
    }
}

extern "C" void kernel_launch(void* const* d_in, const int* in_sizes, int n_in,
                              void* d_out, int out_size, void* d_ws, size_t ws_size,
                              hipStream_t stream) {
    const float* x    = (const float*)d_in[0];   // 8*32*64*64
    const float* w    = (const float*)d_in[1];   // 64*64*32*32*3*3
    const float* bias = (const float*)d_in[2];   // 32
    float* out        = (float*)d_out;           // 8*32*64*64

    // 4096 locations, one wave each, 4 waves per 128-thread block.
    ConvLocal2d_wmma_f32_kernel<<<4096 / WAVES_PER_BLOCK, 128, 0, stream>>>(x, w, bias, out);
}